// OneHotProjected_54881092108375
// MI455X (gfx1250) — compile-verified
//
#include <hip/hip_runtime.h>
#include <cstdint>

// out[tok, :] = W[:, x[tok]]  -- pure column-gather, 0 FLOPs.
// W: [512, 50257] f32 row-major; out: [32768, 512] f32.
// HBM floor ~167 MiB (~7.2 us @ 23.3 TB/s): W streamed into L2 once (103 MiB,
// ~21x cache-line reuse across 32768 random tokens; fits in 192 MB L2) plus
// 64 MiB of non-temporal output stores that don't evict W from L2.
//
// CDNA5 path: per-wave async gather (global_load_async_to_lds_b32, ASYNCcnt)
// of the VOCAB-strided column into contiguous LDS, double-buffered across two
// tokens per wave using the in-order completion guarantee of async loads
// (s_wait_asynccnt 16 releases token A while token B is still in flight),
// then ds_load_b128 + nontemporal global_store_b128 write-out.

#define PROJ      512
#define VOCAB_SZ  50257
#define WAVES_PER_BLOCK 8     // 256 threads, wave32
#define TOKENS_PER_WAVE 2

typedef float v4f __attribute__((ext_vector_type(4)));

__global__ __launch_bounds__(256)
void OneHotProjected_gather_kernel(const int* __restrict__ x,
                                   const float* __restrict__ W,
                                   float* __restrict__ out,
                                   int n_tokens)
{
    // Two 512-f32 column slabs per wave: 8 waves * 2 * 2KB = 32KB LDS / block.
    __shared__ float lds_col[WAVES_PER_BLOCK * TOKENS_PER_WAVE * PROJ];

    const int lane = threadIdx.x & 31;
    const int wave = threadIdx.x >> 5;
    const int tok0 = (blockIdx.x * WAVES_PER_BLOCK + wave) * TOKENS_PER_WAVE;
    if (tok0 >= n_tokens) return;          // uniform per wave

    // Token ids are wave-uniform -> SGPRs.
    const int tA = __builtin_amdgcn_readfirstlane(x[tok0]);
    const int tB = __builtin_amdgcn_readfirstlane(x[tok0 + 1]);

    float* colA = &lds_col[(wave * TOKENS_PER_WAVE + 0) * PROJ];
    float* colB = &lds_col[(wave * TOKENS_PER_WAVE + 1) * PROJ];

    // --- Issue phase: 16 async 4B gathers per lane per token (32 in flight).
    // Each element of the column is VOCAB*4 bytes apart -> one L2 line each,
    // amortized across tokens by L2 residency of W. No VGPR return data.
    #pragma unroll
    for (int i = 0; i < 16; ++i) {
        const int p = i * 32 + lane;                          // proj 0..511
        const float* gA = W + (uint32_t)(p * VOCAB_SZ) + tA;
        uint32_t lA = (uint32_t)(uintptr_t)(colA + p);        // LDS byte addr
        asm volatile("global_load_async_to_lds_b32 %0, %1, off"
                     :: "v"(lA), "v"(gA) : "memory");
    }
    #pragma unroll
    for (int i = 0; i < 16; ++i) {
        const int p = i * 32 + lane;
        const float* gB = W + (uint32_t)(p * VOCAB_SZ) + tB;
        uint32_t lB = (uint32_t)(uintptr_t)(colB + p);
        asm volatile("global_load_async_to_lds_b32 %0, %1, off"
                     :: "v"(lB), "v"(gB) : "memory");
    }

    // --- Drain A while B is still in flight: async loads complete in order,
    // so ASYNCcnt <= 16 implies all 16 of token A's gathers have landed.
    asm volatile("s_wait_asynccnt 0x10" ::: "memory");

    float* dstA = out + (uint64_t)tok0 * PROJ;
    #pragma unroll
    for (int i = 0; i < 4; ++i) {
        const int off = i * 128 + lane * 4;                   // float4 step
        v4f v = *(const v4f*)(colA + off);                    // ds_load_b128
        __builtin_nontemporal_store(v, (v4f*)(dstA + off));   // b128 store NT
    }

    // --- Drain B.
    asm volatile("s_wait_asynccnt 0x0" ::: "memory");

    float* dstB = out + (uint64_t)(tok0 + 1) * PROJ;
    #pragma unroll
    for (int i = 0; i < 4; ++i) {
        const int off = i * 128 + lane * 4;
        v4f v = *(const v4f*)(colB + off);
        __builtin_nontemporal_store(v, (v4f*)(dstB + off));
    }
}

extern "C" void kernel_launch(void* const* d_in, const int* in_sizes, int n_in,
                              void* d_out, int out_size, void* d_ws, size_t ws_size,
                              hipStream_t stream) {
    (void)n_in; (void)out_size; (void)d_ws; (void)ws_size;
    const int*   x = (const int*)d_in[0];     // [B*S] token ids
    const float* W = (const float*)d_in[1];   // [512, 50257] f32
    float*     out = (float*)d_out;           // [B*S, 512] f32

    const int n_tokens = in_sizes[0];         // 8 * 4096 = 32768
    const int toks_per_block = WAVES_PER_BLOCK * TOKENS_PER_WAVE;
    const int blocks = (n_tokens + toks_per_block - 1) / toks_per_block;

    OneHotProjected_gather_kernel<<<blocks, 256, 0, stream>>>(x, W, out, n_tokens);
}